// Model_27659589386541
// MI455X (gfx1250) — compile-verified
//
#include <hip/hip_runtime.h>
#include <cstddef>

typedef __attribute__((ext_vector_type(2))) float v2f;
typedef __attribute__((ext_vector_type(4))) float v4f;
typedef __attribute__((ext_vector_type(8))) float v8f;

static inline int ceil_div(int a, int b) { return (a + b - 1) / b; }

// ---------------------------------------------------------------------------
// WMMA f32 GEMM: C[M,N] = op(A[M,K] @ B) (+bias)(+C)(relu)
//   TRANSB==false : B is [K,N] row-major (ldb = row stride)
//   TRANSB==true  : B is [N,K] row-major (ldb = row stride), i.e. C = A @ B^T
// Block = 8 waves = 128 rows x 128 cols of C. B slabs (32K x 128N) are staged
// through double-buffered LDS in a k-pair-interleaved layout:
//   Bl[k/2][c*2 + (k&1)]  (row stride PR=288 dwords, banks split per k-parity)
// so each 16x16 tile's B fragment is ONE ds_load_b64 (compiler merges pairs
// into ds_load_2addr_b64). A fragments for the whole slab are prefetched into
// registers at slab top, overlapping all global latency with 64 WMMAs/slab.
// The slab body is specialized on steps==8 vs 4 so the common path has no
// per-step guards (single uniform branch per slab).
// ---------------------------------------------------------------------------
template <bool TRANSB>
__global__ __launch_bounds__(256) void wmma_gemm_f32(
    const float* __restrict__ A, const float* __restrict__ B,
    float* __restrict__ C, int M, int N, int K, int lda, int ldb, int ldc,
    const float* __restrict__ bias, int accum, int relu) {
  constexpr int PR = 288;                      // dwords per k-pair row
  __shared__ float Bl[2][16 * PR];             // 36.9 KB total

  const int tid = threadIdx.x;
  const int wave = tid >> 5;                   // wave32
  const int lane = tid & 31;
  const int mt = blockIdx.y * 8 + wave;        // 16-row tile index
  const bool mvalid = (mt * 16) < M;           // no early return: barriers!
  const int row0 = mvalid ? mt * 16 : 0;       // clamp for safe loads
  const int colBase = blockIdx.x * 128;
  const int l15 = lane & 15;
  const int khi = (lane >> 4) << 1;            // 0 or 2 (A/B K sub-offset)

  const float* Ap = A + (size_t)(row0 + l15) * (size_t)lda + khi;

  // --- staging thread mappings ---
  // NN: unit = 2k x 2c; thread t -> cpair = t&63, kp0 = t>>6, kp = kp0 + 4j
  const int nnCpair = tid & 63;
  const int nnKp0 = tid >> 6;
  // NT: thread t -> one col, 4 k-quads at half*16 + j*4
  const int ntCol = tid & 127;
  const int ntColC = (colBase + ntCol) < N ? (colBase + ntCol) : (N - 1);
  const int ntHalf = tid >> 7;
  const v4f vzero = {};

  v4f rr[4];
  auto gload = [&](int kb, int kbs2) {
#pragma unroll
    for (int j = 0; j < 4; ++j) {
      if (TRANSB) {
        int kk = ntHalf * 16 + j * 4;          // uniform per wave
        rr[j] = (kk < kbs2) ? *(const v4f*)(B + (size_t)ntColC * ldb + kb + kk)
                            : vzero;
      } else {
        int kp = nnKp0 + 4 * j;                // uniform per wave
        if (2 * kp < kbs2) {
          const float* p = B + (size_t)(kb + 2 * kp) * ldb + colBase + nnCpair * 2;
          v2f r0 = *(const v2f*)p;             // B[2kp  ][c..c+1]
          v2f r1 = *(const v2f*)(p + ldb);     // B[2kp+1][c..c+1]
          v4f q;
          q[0] = r0.x; q[1] = r1.x; q[2] = r0.y; q[3] = r1.y;
          rr[j] = q;
        } else {
          rr[j] = vzero;
        }
      }
    }
  };
  auto lstore = [&](int bf) {
#pragma unroll
    for (int j = 0; j < 4; ++j) {
      if (TRANSB) {
        int kp = (ntHalf * 16 + j * 4) >> 1;   // even
        v2f p0, p1;
        p0.x = rr[j][0]; p0.y = rr[j][1];
        p1.x = rr[j][2]; p1.y = rr[j][3];
        *(v2f*)&Bl[bf][kp * PR + ntCol * 2] = p0;
        *(v2f*)&Bl[bf][(kp + 1) * PR + ntCol * 2] = p1;
      } else {
        int kp = nnKp0 + 4 * j;
        *(v4f*)&Bl[bf][kp * PR + nnCpair * 4] = rr[j];
      }
    }
  };

  v8f acc[8] = {};
  const int ibase = (khi >> 1) * PR + l15 * 2; // per-lane LDS read base

  int kbs = K < 32 ? K : 32;
  gload(0, kbs);
  lstore(0);
  int buf = 0;

  for (int kb = 0; kb < K; kb += 32) {
    kbs = (K - kb) < 32 ? (K - kb) : 32;       // uniform across block
    const int steps = kbs >> 2;                // uniform (4 or 8)
    __syncthreads();                           // Bl[buf] visible to all

    const int kbNext = kb + 32;
    const bool hasNext = kbNext < K;           // uniform
    const int kbsNext = (K - kbNext) < 32 ? (K - kbNext) : 32;

    v2f av[8];
    auto step = [&](int s) {
      const float* blp = &Bl[buf][ibase + s * 2 * PR];
      v2f b[8];
#pragma unroll
      for (int t = 0; t < 8; ++t)              // ds_load_2addr_b64 pairs
        b[t] = *(const v2f*)(blp + t * 32);
#pragma unroll
      for (int t = 0; t < 8; ++t)
        acc[t] = __builtin_amdgcn_wmma_f32_16x16x4_f32(
            false, av[s], false, b[t], (short)0, acc[t], false, false);
    };

    if (steps == 8) {                          // common path: no per-step guards
#pragma unroll
      for (int s = 0; s < 8; ++s) av[s] = *(const v2f*)(Ap + (s << 2));
      Ap += 32;
      if (hasNext) gload(kbNext, kbsNext);
#pragma unroll
      for (int s = 0; s < 8; ++s) step(s);
    } else {                                   // K-tail slab (steps == 4)
#pragma unroll
      for (int s = 0; s < 4; ++s) av[s] = *(const v2f*)(Ap + (s << 2));
      Ap += 16;
      if (hasNext) gload(kbNext, kbsNext);
#pragma unroll
      for (int s = 0; s < 4; ++s) step(s);
    }

    if (hasNext) lstore(buf ^ 1);              // prev readers passed barrier
    buf ^= 1;
  }

  if (!mvalid) return;                         // after all barriers
  const int rofs = (lane >> 4) << 3;           // lanes 16-31 hold rows +8
#pragma unroll
  for (int t = 0; t < 8; ++t) {
    int col = colBase + t * 16 + l15;
    if (col >= N) continue;                    // whole tile OOB (N % 16 == 0)
#pragma unroll
    for (int rw = 0; rw < 8; ++rw) {
      int row = row0 + rofs + rw;
      size_t idx = (size_t)row * ldc + col;
      float v = acc[t][rw];
      if (bias) v += bias[col];
      if (accum) v += C[idx];
      if (relu) v = v > 0.f ? v : 0.f;
      C[idx] = v;
    }
  }
}

// ------------------------------ elementwise --------------------------------
__global__ void fill_kernel(float* p, float v, long n) {
  long i = (long)blockIdx.x * blockDim.x + threadIdx.x;
  if (i < n) p[i] = v;
}

__global__ void deg_kernel(const int* __restrict__ dst, int E, float* deg) {
  int i = blockIdx.x * blockDim.x + threadIdx.x;
  if (i < E) atomicAdd(&deg[dst[i]], 1.0f);
}

__global__ void rsqrt_kernel(float* d, int n) {
  int i = blockIdx.x * blockDim.x + threadIdx.x;
  if (i < n) d[i] = rsqrtf(d[i]);
}

// GCN edge scatter: agg[dst] += h[src] * dinv[src]*dinv[dst]
__global__ void gcn_scatter(const float* __restrict__ h, const int* __restrict__ src,
                            const int* __restrict__ dst, const float* __restrict__ dinv,
                            float* agg, long E) {
  long idx = (long)blockIdx.x * blockDim.x + threadIdx.x;
  if (idx >= E * 128) return;
  int e = (int)(idx >> 7), c = (int)(idx & 127);
  int s = src[e], d = dst[e];
  atomicAdd(&agg[(size_t)d * 128 + c], h[(size_t)s * 128 + c] * dinv[s] * dinv[d]);
}

// out = relu(agg + self_loop + bias + residual)
__global__ void gcn_finalize(const float* __restrict__ agg, const float* __restrict__ h,
                             const float* __restrict__ dinv, const float* __restrict__ bias,
                             const float* __restrict__ resid, float* out, long n) {
  long idx = (long)blockIdx.x * blockDim.x + threadIdx.x;
  if (idx >= n * 128) return;
  int i = (int)(idx >> 7), c = (int)(idx & 127);
  float di = dinv[i];
  float v = agg[idx] + h[idx] * di * di + bias[c] + resid[idx];
  out[idx] = v > 0.f ? v : 0.f;
}

// per-node, per-head attention projections: as_[i,h]=<hgat[i,h,:],a_src[h,:]>
__global__ void att_coeff(const float* __restrict__ hg, const float* __restrict__ a_src,
                          const float* __restrict__ a_dst, float* as_, float* ad_, int n) {
  int idx = blockIdx.x * blockDim.x + threadIdx.x;
  if (idx >= n * 4) return;
  int i = idx >> 2, hd = idx & 3;
  const float* hp = hg + (size_t)i * 512 + hd * 128;
  const float* sp = a_src + hd * 128;
  const float* dp = a_dst + hd * 128;
  float vs = 0.f, vd = 0.f;
  for (int c = 0; c < 128; ++c) { float x = hp[c]; vs += x * sp[c]; vd += x * dp[c]; }
  as_[idx] = vs;
  ad_[idx] = vd;
}

__device__ inline void atomicMaxFloat(float* addr, float value) {
  if (value >= 0.f)
    atomicMax((int*)addr, __float_as_int(value));
  else
    atomicMin((unsigned int*)addr, __float_as_uint(value));
}

// edges [0,E) real; [E,E+n) self loops
__global__ void gat_edge_logits(const float* __restrict__ as_, const float* __restrict__ ad_,
                                const int* __restrict__ src, const int* __restrict__ dst,
                                float* el, float* emax, int E, int n) {
  int e = blockIdx.x * blockDim.x + threadIdx.x;
  if (e >= E + n) return;
  int s = e < E ? src[e] : (e - E);
  int d = e < E ? dst[e] : (e - E);
#pragma unroll
  for (int hd = 0; hd < 4; ++hd) {
    float v = as_[s * 4 + hd] + ad_[d * 4 + hd];
    v = v > 0.f ? v : 0.2f * v;                // leaky_relu(0.2)
    el[(size_t)e * 4 + hd] = v;
    atomicMaxFloat(&emax[d * 4 + hd], v);
  }
}

__global__ void gat_edge_exp(float* el, const float* __restrict__ emax,
                             const int* __restrict__ dst, float* denom, int E, int n) {
  int e = blockIdx.x * blockDim.x + threadIdx.x;
  if (e >= E + n) return;
  int d = e < E ? dst[e] : (e - E);
#pragma unroll
  for (int hd = 0; hd < 4; ++hd) {
    float ex = expf(el[(size_t)e * 4 + hd] - emax[d * 4 + hd]);
    el[(size_t)e * 4 + hd] = ex;
    atomicAdd(&denom[d * 4 + hd], ex);
  }
}

// agg[dst,c] += 0.25 * sum_h hgat[src,h,c] * ex/denom
__global__ void gat_scatter(const float* __restrict__ hg, const float* __restrict__ ex,
                            const float* __restrict__ denom, const int* __restrict__ src,
                            const int* __restrict__ dst, float* agg, int E, int n) {
  long idx = (long)blockIdx.x * blockDim.x + threadIdx.x;
  long tot = (long)(E + n) * 128;
  if (idx >= tot) return;
  int e = (int)(idx >> 7), c = (int)(idx & 127);
  int s = e < E ? src[e] : (e - E);
  int d = e < E ? dst[e] : (e - E);
  float a = 0.f;
#pragma unroll
  for (int hd = 0; hd < 4; ++hd)
    a += hg[(size_t)s * 512 + hd * 128 + c] * (ex[(size_t)e * 4 + hd] / denom[(size_t)d * 4 + hd]);
  atomicAdd(&agg[(size_t)d * 128 + c], 0.25f * a);
}

__global__ void gat_finalize(const float* __restrict__ agg, const float* __restrict__ bias,
                             const float* __restrict__ resid, float* out, long n) {
  long idx = (long)blockIdx.x * blockDim.x + threadIdx.x;
  if (idx >= n * 128) return;
  int c = (int)(idx & 127);
  float v = agg[idx] + bias[c] + resid[idx];
  out[idx] = v > 0.f ? v : 0.f;
}

__global__ void concat_kernel(const float* __restrict__ ehet, const float* __restrict__ eown,
                              float* out, int n, int hetOff) {
  long idx = (long)blockIdx.x * blockDim.x + threadIdx.x;
  if (idx >= (long)n * 256) return;
  int i = (int)(idx >> 8), c = (int)(idx & 255);
  out[idx] = (c < 128) ? ehet[(size_t)(i + hetOff) * 128 + c]
                       : eown[(size_t)i * 128 + (c - 128)];
}

// ------------------------------ host driver --------------------------------
static void launch_gemm(const float* A, const float* B, float* C, int M, int N, int K,
                        int lda, int ldb, int ldc, int transB, const float* bias,
                        int accum, int relu, hipStream_t s) {
  dim3 grid((unsigned)ceil_div(N, 128), (unsigned)ceil_div(M / 16, 8));
  if (transB)
    wmma_gemm_f32<true><<<grid, 256, 0, s>>>(A, B, C, M, N, K, lda, ldb, ldc, bias, accum, relu);
  else
    wmma_gemm_f32<false><<<grid, 256, 0, s>>>(A, B, C, M, N, K, lda, ldb, ldc, bias, accum, relu);
}

struct BranchParams {
  const float *pW, *pb, *g1W, *g1b, *aW, *as, *ad, *ab, *g2W, *g2b, *cW, *cb;
};

static void run_branch(const float* x, const int* src, const int* dst, int E, int n, int din,
                       const BranchParams& bp, float* emb, float* proj, float* hbuf, float* agg,
                       float* g1, float* ga, float* dinv, float* as_, float* ad_, float* emax,
                       float* denom, float* el, hipStream_t s) {
  const int T = 256;
  const long n128 = (long)n * 128;
  // projection: proj = x @ pW + pb
  launch_gemm(x, bp.pW, proj, n, 128, din, din, 128, 128, 0, bp.pb, 0, 0, s);
  // symmetric-norm degrees (self loops -> init 1)
  fill_kernel<<<ceil_div(n, T), T, 0, s>>>(dinv, 1.0f, n);
  deg_kernel<<<ceil_div(E, T), T, 0, s>>>(dst, E, dinv);
  rsqrt_kernel<<<ceil_div(n, T), T, 0, s>>>(dinv, n);
  // GCN1: g1 = relu(Agg(proj @ g1W) + g1b + proj)
  launch_gemm(proj, bp.g1W, hbuf, n, 128, 128, 128, 128, 128, 0, nullptr, 0, 0, s);
  fill_kernel<<<(int)((n128 + T - 1) / T), T, 0, s>>>(agg, 0.f, n128);
  gcn_scatter<<<(int)(((long)E * 128 + T - 1) / T), T, 0, s>>>(hbuf, src, dst, dinv, agg, E);
  gcn_finalize<<<(int)((n128 + T - 1) / T), T, 0, s>>>(agg, hbuf, dinv, bp.g1b, proj, g1, n);
  // GAT: ga = relu(mean_h softmax-agg(g1 @ aW) + ab + g1)
  launch_gemm(g1, bp.aW, hbuf, n, 512, 128, 128, 512, 512, 0, nullptr, 0, 0, s);
  att_coeff<<<ceil_div(n * 4, T), T, 0, s>>>(hbuf, bp.as, bp.ad, as_, ad_, n);
  fill_kernel<<<ceil_div(n * 4, T), T, 0, s>>>(emax, -3.0e38f, (long)n * 4);
  fill_kernel<<<ceil_div(n * 4, T), T, 0, s>>>(denom, 0.f, (long)n * 4);
  gat_edge_logits<<<ceil_div(E + n, T), T, 0, s>>>(as_, ad_, src, dst, el, emax, E, n);
  gat_edge_exp<<<ceil_div(E + n, T), T, 0, s>>>(el, emax, dst, denom, E, n);
  fill_kernel<<<(int)((n128 + T - 1) / T), T, 0, s>>>(agg, 0.f, n128);
  gat_scatter<<<(int)(((long)(E + n) * 128 + T - 1) / T), T, 0, s>>>(hbuf, el, denom, src, dst, agg, E, n);
  gat_finalize<<<(int)((n128 + T - 1) / T), T, 0, s>>>(agg, bp.ab, g1, ga, n);
  // GCN2: g2 (reuse proj) = relu(Agg(ga @ g2W) + g2b + ga)
  launch_gemm(ga, bp.g2W, hbuf, n, 128, 128, 128, 128, 128, 0, nullptr, 0, 0, s);
  fill_kernel<<<(int)((n128 + T - 1) / T), T, 0, s>>>(agg, 0.f, n128);
  gcn_scatter<<<(int)(((long)E * 128 + T - 1) / T), T, 0, s>>>(hbuf, src, dst, dinv, agg, E);
  gcn_finalize<<<(int)((n128 + T - 1) / T), T, 0, s>>>(agg, hbuf, dinv, bp.g2b, ga, proj, n);
  // conv einsum: emb = g1 @ cW[:,0,:]^T + g2 @ cW[:,1,:]^T + cb
  launch_gemm(g1, bp.cW, emb, n, 128, 128, 128, 256, 128, 1, nullptr, 0, 0, s);
  launch_gemm(proj, bp.cW + 128, emb, n, 128, 128, 128, 256, 128, 1, bp.cb, 1, 0, s);
}

extern "C" void kernel_launch(void* const* d_in, const int* in_sizes, int n_in,
                              void* d_out, int out_size, void* d_ws, size_t ws_size,
                              hipStream_t stream) {
  const int HERBN = 4000, TGTN = 6000, HETN = 10000;
  const int E_HET = 200000, E_HERB = 80000, E_TGT = 120000;
  (void)in_sizes; (void)n_in; (void)out_size; (void)ws_size;

  const int* het_net = (const int*)d_in[0];
  const float* het_x = (const float*)d_in[1];
  const int* herb_net = (const int*)d_in[2];
  const float* herb_x = (const float*)d_in[3];
  const int* tgt_net = (const int*)d_in[4];
  const float* tgt_x = (const float*)d_in[5];

  const float* P[36];
  for (int j = 0; j < 36; ++j) P[j] = (const float*)d_in[6 + j];
  auto mk = [&](int b) {
    BranchParams bp; const float* const* q = P + b * 12;
    bp.pW = q[0]; bp.pb = q[1]; bp.g1W = q[2]; bp.g1b = q[3];
    bp.aW = q[4]; bp.as = q[5]; bp.ad = q[6]; bp.ab = q[7];
    bp.g2W = q[8]; bp.g2b = q[9]; bp.cW = q[10]; bp.cb = q[11];
    return bp;
  };
  BranchParams bph = mk(0), bpb = mk(1), bpt = mk(2);

  // workspace carve (floats), sized for the largest branch (het)
  const size_t NMAX = 10000, EMAX = 210000;
  float* ws = (float*)d_ws;
  size_t off = 0;
  auto alloc = [&](size_t ne) { float* p = ws + off; off += ne; return p; };
  float* proj = alloc(NMAX * 128);
  float* hbuf = alloc(NMAX * 512);
  float* agg  = alloc(NMAX * 128);
  float* g1   = alloc(NMAX * 128);
  float* ga   = alloc(NMAX * 128);
  float* dinv = alloc(NMAX);
  float* as_  = alloc(NMAX * 4);
  float* ad_  = alloc(NMAX * 4);
  float* emax = alloc(NMAX * 4);
  float* denom = alloc(NMAX * 4);
  float* el   = alloc(EMAX * 4);
  float* ehet = alloc((size_t)HETN * 128);
  float* eherb = alloc((size_t)HERBN * 128);
  float* etgt = alloc((size_t)TGTN * 128);
  float* catH = alloc((size_t)HERBN * 256);
  float* catT = alloc((size_t)TGTN * 256);

  run_branch(het_x, het_net, het_net + E_HET, E_HET, HETN, HETN, bph, ehet,
             proj, hbuf, agg, g1, ga, dinv, as_, ad_, emax, denom, el, stream);
  run_branch(herb_x, herb_net, herb_net + E_HERB, E_HERB, HERBN, HERBN, bpb, eherb,
             proj, hbuf, agg, g1, ga, dinv, as_, ad_, emax, denom, el, stream);
  run_branch(tgt_x, tgt_net, tgt_net + E_TGT, E_TGT, TGTN, TGTN, bpt, etgt,
             proj, hbuf, agg, g1, ga, dinv, as_, ad_, emax, denom, el, stream);

  concat_kernel<<<ceil_div(HERBN * 256, 256), 256, 0, stream>>>(ehet, eherb, catH, HERBN, 0);
  concat_kernel<<<ceil_div(TGTN * 256, 256), 256, 0, stream>>>(ehet, etgt, catT, TGTN, HERBN);

  // out = catH @ catT^T   [4000 x 6000]
  launch_gemm(catH, catT, (float*)d_out, HERBN, TGTN, 256, 256, 256, TGTN, 1,
              nullptr, 0, 0, stream);
}